// BatchGraphConv_82815559402091
// MI455X (gfx1250) — compile-verified
//
#include <hip/hip_runtime.h>

typedef __attribute__((ext_vector_type(2))) float v2f;
typedef __attribute__((ext_vector_type(8))) float v8f;

#define D 64  // D_IN == D_OUT == 64

// ---------------------------------------------------------------------------
// Kernel 1: h[N,64] = x[N,64] @ W[64,64] + b   (fp32 WMMA 16x16x4)
// One wave32 computes a 16x64 output strip. 4 col-tiles x 16 K-steps.
//
// Fragment layouts (CDNA5 ISA 7.12.2, 32-bit):
//   A 16x4 : lanes 0-15 row M=l, v0=K0 v1=K1 ; lanes 16-31 row M=l-16, v0=K2 v1=K3
//            -> per lane a contiguous float2 load from x[row, k + 2*half]
//   B 4x16 : lanes 0-15 col N=l, v0=row K0 v1=row K1 ; lanes 16-31 v0=K2 v1=K3
//   C/D    : VGPR v -> (M = v + 8*half, N = lane&15)
// ---------------------------------------------------------------------------
__global__ __launch_bounds__(256) void gcn_gemm_kernel(
    const float* __restrict__ x, const float* __restrict__ W,
    const float* __restrict__ bias, float* __restrict__ h,
    int nRowTiles, int N)
{
  const int lane = threadIdx.x & 31;
  const int tile = blockIdx.x * 8 + (threadIdx.x >> 5);   // wave-uniform
  if (tile >= nRowTiles) return;                          // whole wave exits together

  const int m0   = tile * 16;
  const int half = lane >> 4;      // 0: lanes 0-15, 1: lanes 16-31
  const int l15  = lane & 15;

  int mrow = m0 + l15;
  if (mrow >= N) mrow = N - 1;     // clamp loads; keep EXEC all-ones for WMMA
  const float* __restrict__ xrow = x + (size_t)mrow * D;

  v8f acc0 = {}, acc1 = {}, acc2 = {}, acc3 = {};

  #pragma unroll
  for (int k = 0; k < D; k += 4) {
    // A fragment: contiguous 8B load per lane
    const v2f a = *(const v2f*)(xrow + k + 2 * half);
    const int kr = (k + 2 * half) * D + l15;   // W[k + 2*half][l15 + n0]
    {
      v2f bf; bf.x = W[kr + 0 * D +  0]; bf.y = W[kr + 1 * D +  0];
      acc0 = __builtin_amdgcn_wmma_f32_16x16x4_f32(false, a, false, bf,
                                                   (short)0, acc0, false, false);
    }
    {
      v2f bf; bf.x = W[kr + 0 * D + 16]; bf.y = W[kr + 1 * D + 16];
      acc1 = __builtin_amdgcn_wmma_f32_16x16x4_f32(false, a, false, bf,
                                                   (short)0, acc1, false, false);
    }
    {
      v2f bf; bf.x = W[kr + 0 * D + 32]; bf.y = W[kr + 1 * D + 32];
      acc2 = __builtin_amdgcn_wmma_f32_16x16x4_f32(false, a, false, bf,
                                                   (short)0, acc2, false, false);
    }
    {
      v2f bf; bf.x = W[kr + 0 * D + 48]; bf.y = W[kr + 1 * D + 48];
      acc3 = __builtin_amdgcn_wmma_f32_16x16x4_f32(false, a, false, bf,
                                                   (short)0, acc3, false, false);
    }
  }

  // Store D = acc + bias.  VGPR v holds row (m0 + v + 8*half), col (t*16 + l15).
  v8f accs[4] = {acc0, acc1, acc2, acc3};
  #pragma unroll
  for (int t = 0; t < 4; ++t) {
    const int col = t * 16 + l15;
    const float bv = bias[col];
    #pragma unroll
    for (int v = 0; v < 8; ++v) {
      const int row = m0 + v + 8 * half;
      if (row < N) h[(size_t)row * D + col] = accs[t][v] + bv;
    }
  }
}

// ---------------------------------------------------------------------------
// Kernel 2: zero-init out (harness poisons d_out; atomics need zeros)
// ---------------------------------------------------------------------------
__global__ __launch_bounds__(256) void gcn_zero_kernel(float4* __restrict__ out, int n4)
{
  const int i = blockIdx.x * blockDim.x + threadIdx.x;
  if (i < n4) out[i] = make_float4(0.f, 0.f, 0.f, 0.f);
}

// ---------------------------------------------------------------------------
// Kernel 3: edge scatter. One wave32 per edge; lane owns 2 features.
// Gather of h rows is fully coalesced b64 and L2-resident (h = 25.6 MB << 192 MB L2).
// unsafeAtomicAdd -> hardware global_atomic_add_f32 (no CAS loop), resolved at L2.
// ---------------------------------------------------------------------------
__global__ __launch_bounds__(256) void gcn_scatter_kernel(
    const float* __restrict__ h, const float* __restrict__ adj,
    const int* __restrict__ esrc, const int* __restrict__ edst,
    float* __restrict__ out, int E)
{
  const int lane = threadIdx.x & 31;
  const int e = blockIdx.x * 8 + (threadIdx.x >> 5);
  if (e >= E) return;

  const float a = adj[e];          // same addr across wave -> broadcast from cache
  const int   s = esrc[e];
  const int   d = edst[e];

  const v2f hv = *(const v2f*)(h + (size_t)s * D + 2 * lane);
  float* dst = out + (size_t)d * D + 2 * lane;
  unsafeAtomicAdd(dst + 0, a * hv.x);
  unsafeAtomicAdd(dst + 1, a * hv.y);
}

// ---------------------------------------------------------------------------
// Kernel 4: in-place ReLU on out
// ---------------------------------------------------------------------------
__global__ __launch_bounds__(256) void gcn_relu_kernel(float4* __restrict__ out, int n4)
{
  const int i = blockIdx.x * blockDim.x + threadIdx.x;
  if (i < n4) {
    float4 v = out[i];
    v.x = fmaxf(v.x, 0.f); v.y = fmaxf(v.y, 0.f);
    v.z = fmaxf(v.z, 0.f); v.w = fmaxf(v.w, 0.f);
    out[i] = v;
  }
}

extern "C" void kernel_launch(void* const* d_in, const int* in_sizes, int n_in,
                              void* d_out, int out_size, void* d_ws, size_t ws_size,
                              hipStream_t stream)
{
  // setup_inputs() order: x, adj_vals, W, b, edge_src, edge_dst
  const float* x    = (const float*)d_in[0];
  const float* adj  = (const float*)d_in[1];
  const float* W    = (const float*)d_in[2];
  const float* bias = (const float*)d_in[3];
  const int*   esrc = (const int*)d_in[4];
  const int*   edst = (const int*)d_in[5];
  float* out = (float*)d_out;
  float* h   = (float*)d_ws;                 // N*64 floats = 25.6 MB scratch

  const int N = in_sizes[0] / D;
  const int E = in_sizes[1];

  const int nRowTiles = (N + 15) / 16;
  gcn_gemm_kernel<<<(nRowTiles + 7) / 8, 256, 0, stream>>>(x, W, bias, h, nRowTiles, N);

  const int n4 = (N * D) / 4;
  gcn_zero_kernel<<<(n4 + 255) / 256, 256, 0, stream>>>((float4*)out, n4);

  gcn_scatter_kernel<<<(E + 7) / 8, 256, 0, stream>>>(h, adj, esrc, edst, out, E);

  gcn_relu_kernel<<<(n4 + 255) / 256, 256, 0, stream>>>((float4*)out, n4);
}